// split_linear_15925738733998
// MI455X (gfx1250) — compile-verified
//
#include <hip/hip_runtime.h>

// ---------------------------------------------------------------------------
// split-linear: out[b,d,g] = sum_f x[b,d,f] * W[d,g,f] + bias[d,g]
//   x: (2048, 512, 64) f32   W: (512, 64, 64) f32   bias: (512, 64) f32
//   out: (2048, 512, 64) f32
//
// MI455X plan:
//   - memory-bound (~16 flop/byte -> ~23us floor @ 23.3 TB/s); fp32 WMMA
//     (V_WMMA_F32_16X16X4_F32) keeps compute below that floor and is exact.
//   - WG = 256 thr = 8 waves handles (d, 128 batch rows).
//   - Staging via Tensor Data Mover: one TENSOR_LOAD_TO_LDS descriptor for the
//     X tile (128x64, row stride 32768 elems) and one for W_d (64x64), with
//     D# LDS padding (64 DWORDs + 4 DWORD pad) reproducing the 68-float LDS
//     stride -> 16B-aligned rows, (4*row+col)%64 banking, conflict-free frags.
//   - Each wave: 16 rows x 64 cols = 4 accumulators, K=64 in 4-steps
//     -> 64 v_wmma_f32_16x16x4_f32 per wave.
// ---------------------------------------------------------------------------

#define DIM  512
#define FEAT 64
#define BATCH 2048

#define ROWS_PER_WG 128
#define LDS_STRIDE  68   // 64 data floats + 4 pad floats per row

typedef float v2f  __attribute__((ext_vector_type(2)));
typedef float v8f  __attribute__((ext_vector_type(8)));
typedef unsigned int u32x4 __attribute__((ext_vector_type(4)));
typedef int   i32x4 __attribute__((ext_vector_type(4)));
typedef int   i32x8 __attribute__((ext_vector_type(8)));

// Low 32 bits of a flat pointer to LDS == the wave-relative LDS byte address
// (flat->LDS aperture translation truncates to addr[31:0]).
__device__ __forceinline__ unsigned lds_byte_addr(const void* p) {
    return (unsigned)(unsigned long long)p;
}

// Issue one TDM 2-D tile load: tile_d1 rows of tile_d0 f32 elements, global
// row stride `stride0` elements, into LDS at lds_addr with +4-DWORD row pad.
__device__ __forceinline__ void tdm_load_2d_f32(
    unsigned lds_addr, const float* gptr,
    unsigned tensor_d0, unsigned tensor_d1,
    unsigned tile_d0,   unsigned tile_d1,
    unsigned stride0)
{
    const unsigned long long ga = (unsigned long long)gptr;

    // D# group 0 (128b): [1:0] count=1 | [63:32] lds_addr |
    //                    [120:64] global_addr | [127:126] type=2
    u32x4 g0;
    g0.x = 0x1u;
    g0.y = lds_addr;
    g0.z = (unsigned)(ga & 0xFFFFFFFFu);
    g0.w = (unsigned)((ga >> 32) & 0x01FFFFFFu) | (2u << 30);

    // D# group 1 (256b):
    //  [15:0] wg_mask=0  [17:16] data_size=2 (4B)  [20] pad_enable=1
    //  [24:22] pad_interval=5 (64 DW)  [31:25] pad_amount=3 (4 DW)
    //  [79:48] tensor_dim0  [111:80] tensor_dim1
    //  [127:112] tile_dim0  [143:128] tile_dim1  [159:144] tile_dim2=0
    //  [207:160] tensor_dim0_stride  [255:208] tensor_dim1_stride=0
    i32x8 g1;
    g1[0] = (2 << 16) | (1 << 20) | (5 << 22) | (3 << 25);
    g1[1] = (int)(tensor_d0 << 16);                          // dim0[15:0]
    g1[2] = (int)((tensor_d0 >> 16) | (tensor_d1 << 16));    // dim0[31:16]|dim1[15:0]
    g1[3] = (int)((tensor_d1 >> 16) | (tile_d0 << 16));      // dim1[31:16]|tile0
    g1[4] = (int)tile_d1;                                    // tile1 | tile2=0
    g1[5] = (int)stride0;                                    // stride0[31:0]
    g1[6] = 0;                                               // stride0[47:32]|...
    g1[7] = 0;

    const i32x4 gz4 = {0, 0, 0, 0};                          // groups 2/3: 2-D tile
    const i32x8 gz8 = {0, 0, 0, 0, 0, 0, 0, 0};              // extra group (clang-23 form)
    __builtin_amdgcn_tensor_load_to_lds(g0, g1, gz4, gz4, gz8, 0);
}

__global__ __launch_bounds__(256)
void split_linear_wmma_f32_kernel(const float* __restrict__ x,
                                  const float* __restrict__ W,
                                  const float* __restrict__ bias,
                                  float* __restrict__ out)
{
    __shared__ float sX[ROWS_PER_WG * LDS_STRIDE]; // 128 x 64 tile of x (+pad)
    __shared__ float sW[FEAT * LDS_STRIDE];        // 64 x 64 W_d (+pad)
    __shared__ float sB[FEAT];

    const int d       = blockIdx.y;           // 0..511
    const int rowBase = blockIdx.x * ROWS_PER_WG;
    const int tid     = threadIdx.x;          // 0..255
    const int lane    = tid & 31;
    const int wave    = tid >> 5;             // 0..7

    // ---- stage X tile + W_d via Tensor Data Mover (wave 0 issues; TENSORcnt)
    if (wave == 0) {
        tdm_load_2d_f32(lds_byte_addr(sX),
                        x + (long)rowBase * (DIM * FEAT) + (long)d * FEAT,
                        /*tensor_d0=*/FEAT, /*tensor_d1=*/ROWS_PER_WG,
                        /*tile_d0=*/FEAT,   /*tile_d1=*/ROWS_PER_WG,
                        /*stride0=*/DIM * FEAT);
        tdm_load_2d_f32(lds_byte_addr(sW),
                        W + (long)d * FEAT * FEAT,
                        FEAT, FEAT, FEAT, FEAT, FEAT);
        __builtin_amdgcn_s_wait_tensorcnt(0);
    }
    if (tid < FEAT) sB[tid] = bias[(long)d * FEAT + tid];

    __syncthreads();

    // ---- per-wave WMMA: 16 batch rows x 64 output cols
    // A 16x4 f32 layout (ISA 7.12.2): lane 0-15: M=lane, v0=K+0,v1=K+1;
    // lane 16-31: M=lane-16, v0=K+2,v1=K+3. B mirrored; B[k][n] = W_d[n][k].
    const int half = lane >> 4;        // 0 | 1
    const int l16  = lane & 15;
    const int kOff = half << 1;        // 0 | 2
    const int aRow = wave * 16 + l16;

    v8f acc[4];
    #pragma unroll
    for (int nt = 0; nt < 4; ++nt) acc[nt] = (v8f){};

    #pragma unroll
    for (int k = 0; k < FEAT; k += 4) {
        const v2f a = *(const v2f*)(&sX[aRow * LDS_STRIDE + k + kOff]);
        #pragma unroll
        for (int nt = 0; nt < 4; ++nt) {
            const v2f b = *(const v2f*)(&sW[(nt * 16 + l16) * LDS_STRIDE + k + kOff]);
            acc[nt] = __builtin_amdgcn_wmma_f32_16x16x4_f32(
                /*neg_a=*/false, a, /*neg_b=*/false, b,
                /*c_mod=*/(short)0, acc[nt],
                /*reuse_a=*/false, /*reuse_b=*/false);
        }
    }

    // ---- epilogue: bias add + store
    // C/D layout: VGPR r -> row M = r + 8*half ; column N = l16 (all 8 VGPRs)
    const int mBase = rowBase + wave * 16 + half * 8;
    #pragma unroll
    for (int nt = 0; nt < 4; ++nt) {
        const int   col = nt * 16 + l16;
        const float bv  = sB[col];
        #pragma unroll
        for (int r = 0; r < 8; ++r) {
            const long gRow = mBase + r;
            out[gRow * (long)(DIM * FEAT) + (long)d * FEAT + col] = acc[nt][r] + bv;
        }
    }
}

extern "C" void kernel_launch(void* const* d_in, const int* in_sizes, int n_in,
                              void* d_out, int out_size, void* d_ws, size_t ws_size,
                              hipStream_t stream)
{
    const float* x    = (const float*)d_in[0];
    const float* W    = (const float*)d_in[1];
    const float* bias = (const float*)d_in[2];
    float*       out  = (float*)d_out;

    dim3 grid(BATCH / ROWS_PER_WG, DIM);   // (16, 512)
    dim3 block(256);
    split_linear_wmma_f32_kernel<<<grid, block, 0, stream>>>(x, W, bias, out);
}